// GatFeatDecoder_38620345925668
// MI455X (gfx1250) — compile-verified
//
#include <hip/hip_runtime.h>
#include <hip/hip_bf16.h>

typedef __attribute__((ext_vector_type(2))) float v2f;
typedef __attribute__((ext_vector_type(8))) float v8f;

#define ALPHA 0.2f

// ---------------------------------------------------------------------------
// Kernel A: LR[b, i, n] = sum_w V_b[i,w] * Wcat[w,n]
//   V_b[i,w] = x[b,w,i];  Wcat[w,n] = n<128 ? lin_w[n,w] : lin_w[n-128, 64+w]
// One wave per 16x16 output tile. M=256 (i), N=256 (n), Kdim=64 (w).
// ---------------------------------------------------------------------------
__global__ __launch_bounds__(128) void k_linproj(const float* __restrict__ x,
                                                 const float* __restrict__ lin_w,
                                                 float* __restrict__ LR) {
  const int wave = blockIdx.x * 4 + (threadIdx.x >> 5);
  const int lane = threadIdx.x & 31;
  const int half = lane >> 4;
  const int l15  = lane & 15;

  const int tn = wave & 15;          // N tile (256/16)
  const int ti = (wave >> 4) & 15;   // M tile (256/16)
  const int b  = wave >> 8;          // batch

  const int n = tn * 16 + l15;
  // column of Wcat for this lane's B elements
  const float* wcol = (n < 128) ? (lin_w + (size_t)n * 128)
                                : (lin_w + (size_t)(n - 128) * 128 + 64);
  // A: row i0+l15 of V_b, strided reads from x
  const float* acol = x + (size_t)b * 16384 + (ti * 16 + l15);

  v8f c = {0.f, 0.f, 0.f, 0.f, 0.f, 0.f, 0.f, 0.f};
  for (int w0 = 0; w0 < 64; w0 += 4) {
    const int ka = w0 + 2 * half;
    v2f a;
    a.x = acol[(size_t)(ka + 0) * 256];
    a.y = acol[(size_t)(ka + 1) * 256];
    v2f bmat = *(const v2f*)(wcol + ka);     // contiguous, 8B aligned
    c = __builtin_amdgcn_wmma_f32_16x16x4_f32(false, a, false, bmat,
                                              (short)0, c, false, false);
  }

  float* outp = LR + (size_t)b * 65536 + tn * 16 + l15;
#pragma unroll
  for (int r = 0; r < 8; ++r) {
    const int m = r + 8 * half;
    outp[(size_t)(ti * 16 + m) * 256] = c[r];
  }
}

// ---------------------------------------------------------------------------
// Kernel B: attention scores + row softmax -> P[b,i,j]
//   e[j] = sum_e a[e]*lrelu(L[b,i,e] + R[b,j,e] + lin_b[e]) + bias_kk[i,j]
// One 256-thread block per (b, i); thread t handles j = t.
// ---------------------------------------------------------------------------
__global__ __launch_bounds__(256) void k_attn(const float* __restrict__ LR,
                                              const float* __restrict__ lin_b,
                                              const float* __restrict__ a_vec,
                                              const float* __restrict__ bias_kk,
                                              float* __restrict__ P) {
  __shared__ float sL[128];
  __shared__ float sA[128];
  __shared__ float red[8];

  const int b = blockIdx.x >> 8;
  const int i = blockIdx.x & 255;
  const int t = threadIdx.x;

  const float* Lrow = LR + (size_t)b * 65536 + (size_t)i * 256;
  if (t < 128) {
    sL[t] = Lrow[t] + lin_b[t];
    sA[t] = a_vec[t];
  }
  __syncthreads();

  const float* Rrow = LR + (size_t)b * 65536 + (size_t)t * 256 + 128;
  float acc = 0.f;
#pragma unroll 4
  for (int e = 0; e < 128; ++e) {
    float g = sL[e] + Rrow[e];
    g = (g >= 0.f) ? g : ALPHA * g;
    acc = fmaf(sA[e], g, acc);
  }
  acc += bias_kk[(size_t)i * 256 + t];

  // block max (wave32 shuffle + LDS)
  float m = acc;
  for (int off = 16; off; off >>= 1) m = fmaxf(m, __shfl_xor(m, off));
  if ((t & 31) == 0) red[t >> 5] = m;
  __syncthreads();
  float bm = red[0];
#pragma unroll
  for (int w = 1; w < 8; ++w) bm = fmaxf(bm, red[w]);

  const float ex = __expf(acc - bm);

  float s = ex;
  for (int off = 16; off; off >>= 1) s += __shfl_xor(s, off);
  __syncthreads();                        // done reading red[] (max phase)
  if ((t & 31) == 0) red[t >> 5] = s;
  __syncthreads();
  float bs = 0.f;
#pragma unroll
  for (int w = 0; w < 8; ++w) bs += red[w];

  P[(size_t)b * 65536 + (size_t)i * 256 + t] = ex / bs;
}

// ---------------------------------------------------------------------------
// Kernel C: H[b,i,w] = sigmoid( sum_j P[b,i,j] * v[b,j,w] ),  v[b,j,w]=x[b,w,j]
// One wave per 16x16 tile. M=256 (i), N=64 (w), Kdim=256 (j).
// ---------------------------------------------------------------------------
__global__ __launch_bounds__(128) void k_av(const float* __restrict__ P,
                                            const float* __restrict__ x,
                                            float* __restrict__ H) {
  const int wave = blockIdx.x * 4 + (threadIdx.x >> 5);
  const int lane = threadIdx.x & 31;
  const int half = lane >> 4;
  const int l15  = lane & 15;

  const int tn = wave & 3;          // w tile (64/16)
  const int ti = (wave >> 2) & 15;  // i tile (256/16)
  const int b  = wave >> 6;         // batch

  const float* arow = P + (size_t)b * 65536 + (size_t)(ti * 16 + l15) * 256;
  const float* brow = x + (size_t)b * 16384 + (size_t)(tn * 16 + l15) * 256;

  v8f c = {0.f, 0.f, 0.f, 0.f, 0.f, 0.f, 0.f, 0.f};
  for (int k0 = 0; k0 < 256; k0 += 4) {
    const int kk = k0 + 2 * half;
    v2f a    = *(const v2f*)(arow + kk);   // contiguous in j
    v2f bmat = *(const v2f*)(brow + kk);   // contiguous in j
    c = __builtin_amdgcn_wmma_f32_16x16x4_f32(false, a, false, bmat,
                                              (short)0, c, false, false);
  }

  float* outp = H + (size_t)b * 16384 + tn * 16 + l15;
#pragma unroll
  for (int r = 0; r < 8; ++r) {
    const int m = r + 8 * half;
    const float val = c[r];
    outp[(size_t)(ti * 16 + m) * 64] = 1.f / (1.f + __expf(-val));
  }
}

// ---------------------------------------------------------------------------
// Kernel D: out[b,w,o] = sum_k H[b,k,w]*fc_w[o,k] + fc_b[o]
// One wave per 16x16 tile. M=64 (w), N=64 (o), Kdim=256 (k).
// ---------------------------------------------------------------------------
__global__ __launch_bounds__(128) void k_out(const float* __restrict__ H,
                                             const float* __restrict__ fc_w,
                                             const float* __restrict__ fc_b,
                                             float* __restrict__ out) {
  const int wave = blockIdx.x * 4 + (threadIdx.x >> 5);
  const int lane = threadIdx.x & 31;
  const int half = lane >> 4;
  const int l15  = lane & 15;

  const int to = wave & 3;         // o tile
  const int tw = (wave >> 2) & 3;  // w tile
  const int b  = wave >> 4;        // batch

  const float* acol = H + (size_t)b * 16384 + (tw * 16 + l15);   // stride 64 in k
  const float* brow = fc_w + (size_t)(to * 16 + l15) * 256;      // contiguous in k

  v8f c = {0.f, 0.f, 0.f, 0.f, 0.f, 0.f, 0.f, 0.f};
  for (int k0 = 0; k0 < 256; k0 += 4) {
    const int kk = k0 + 2 * half;
    v2f a;
    a.x = acol[(size_t)(kk + 0) * 64];
    a.y = acol[(size_t)(kk + 1) * 64];
    v2f bmat = *(const v2f*)(brow + kk);
    c = __builtin_amdgcn_wmma_f32_16x16x4_f32(false, a, false, bmat,
                                              (short)0, c, false, false);
  }

  const float bias = fc_b[to * 16 + l15];
  float* outp = out + (size_t)b * 4096 + to * 16 + l15;
#pragma unroll
  for (int r = 0; r < 8; ++r) {
    const int m = r + 8 * half;
    outp[(size_t)(tw * 16 + m) * 64] = c[r] + bias;
  }
}

// ---------------------------------------------------------------------------
extern "C" void kernel_launch(void* const* d_in, const int* in_sizes, int n_in,
                              void* d_out, int out_size, void* d_ws, size_t ws_size,
                              hipStream_t stream) {
  (void)in_sizes; (void)n_in; (void)out_size; (void)ws_size;

  const float* x       = (const float*)d_in[0];  // (16,64,256)
  const float* lin_w   = (const float*)d_in[1];  // (128,128)
  const float* lin_b   = (const float*)d_in[2];  // (128)
  const float* a_vec   = (const float*)d_in[3];  // (128)
  const float* bias_kk = (const float*)d_in[4];  // (256,256)
  const float* fc_w    = (const float*)d_in[5];  // (64,256)
  const float* fc_b    = (const float*)d_in[6];  // (64)
  float* out = (float*)d_out;                    // (16,64,64)

  float* LR = (float*)d_ws;            // 16*256*256 = 1,048,576 floats (4 MB)
  float* P  = LR + (size_t)16 * 65536; // 4 MB
  float* H  = P  + (size_t)16 * 65536; // 16*256*64 floats (1 MB)

  // A: 4096 waves -> 1024 blocks x 128 threads
  k_linproj<<<1024, 128, 0, stream>>>(x, lin_w, LR);
  // B: one 256-thread block per (b,i) -> 4096 blocks
  k_attn<<<4096, 256, 0, stream>>>(LR, lin_b, a_vec, bias_kk, P);
  // C: 1024 waves -> 256 blocks x 128
  k_av<<<256, 128, 0, stream>>>(P, x, H);
  // D: 256 waves -> 64 blocks x 128
  k_out<<<64, 128, 0, stream>>>(H, fc_w, fc_b, out);
}